// GIN_26663156973941
// MI455X (gfx1250) — compile-verified
//
#include <hip/hip_runtime.h>
#include <hip/hip_bf16.h>

typedef float v2f __attribute__((ext_vector_type(2)));
typedef float v8f __attribute__((ext_vector_type(8)));

#define DIM 64
#define NGRAPHS 128
#define OUTDIM 10
#define BN_EPS 1e-5f
#define LDS_STRIDE 68   // 64 + 4 pad: conflict-free column reads

// ---------------------------------------------------------------------------
// Copy h -> agg (float4 grid-stride)
// ---------------------------------------------------------------------------
__global__ void gin_copy(const float4* __restrict__ src, float4* __restrict__ dst, long long n4) {
    long long t = (long long)blockIdx.x * blockDim.x + threadIdx.x;
    long long stride = (long long)gridDim.x * blockDim.x;
    for (; t < n4; t += stride) dst[t] = src[t];
}

// ---------------------------------------------------------------------------
// Scatter-add: agg[dst[e]] += h[src[e]].  16 threads/edge, float4 per thread.
// ---------------------------------------------------------------------------
__global__ void gin_scatter(const float* __restrict__ Hin, float* __restrict__ Agg,
                            const int* __restrict__ esrc, const int* __restrict__ edst,
                            long long n_edges) {
    long long t = (long long)blockIdx.x * blockDim.x + threadIdx.x;
    long long e = t >> 4;
    if (e >= n_edges) return;
    int c = (int)(t & 15);            // which float4 chunk of the 64-dim row
    long long s = esrc[e];
    long long d = edst[e];
    const float4 v = *(const float4*)(Hin + s * DIM + c * 4);
    float* dp = Agg + d * DIM + c * 4;
    atomicAdd(dp + 0, v.x);
    atomicAdd(dp + 1, v.y);
    atomicAdd(dp + 2, v.z);
    atomicAdd(dp + 3, v.w);
}

// ---------------------------------------------------------------------------
// Fused GIN MLP, one wave per 64-node supertile (amortizes weight loads 4x):
//   Hout = relu( relu(BN(Hin@W1 + b1)) @ W2 + b2 )   (in-place safe)
// f32 WMMA 16x16x4: K=64 -> 16 k-steps; M=64 -> 4 tiles; N=64 -> 4 tiles.
// 16 accumulators (128 VGPRs), 16 WMMAs per k-step per GEMM.
// ---------------------------------------------------------------------------
__global__ __launch_bounds__(32) void gin_mlp(
    const float* __restrict__ Hin, float* __restrict__ Hout,
    const float* __restrict__ W1, const float* __restrict__ b1,
    const float* __restrict__ gamma, const float* __restrict__ beta,
    const float* __restrict__ rmean, const float* __restrict__ rvar,
    const float* __restrict__ W2, const float* __restrict__ b2,
    int n_nodes) {
    __shared__ float lds[64 * LDS_STRIDE];   // 64x68 f32 = 17.4 KB

    const int lane = threadIdx.x & 31;
    const int m  = lane & 15;     // A row within tile, or B/C/D column
    const int hi = lane >> 4;

    const int row0 = blockIdx.x * 64;

    // Clamped A-row addresses (tail supertile reads duplicate the last row)
    long long arow[4];
    #pragma unroll
    for (int mt = 0; mt < 4; ++mt) {
        int r = row0 + mt * 16 + m;
        arow[mt] = (r < n_nodes) ? r : (long long)(n_nodes - 1);
    }

    v8f acc[4][4];
    #pragma unroll
    for (int mt = 0; mt < 4; ++mt)
        #pragma unroll
        for (int nt = 0; nt < 4; ++nt)
            acc[mt][nt] = (v8f){0,0,0,0,0,0,0,0};

    // ---- GEMM1: acc[mt][nt] = Hin_supertile @ W1 ----
    #pragma unroll
    for (int k = 0; k < DIM; k += 4) {
        const int ka = k + 2 * hi;
        v2f b[4];
        #pragma unroll
        for (int nt = 0; nt < 4; ++nt) {
            b[nt].x = W1[ka * DIM + nt * 16 + m];
            b[nt].y = W1[(ka + 1) * DIM + nt * 16 + m];
        }
        #pragma unroll
        for (int mt = 0; mt < 4; ++mt) {
            v2f a;
            a.x = Hin[arow[mt] * DIM + ka];
            a.y = Hin[arow[mt] * DIM + ka + 1];
            #pragma unroll
            for (int nt = 0; nt < 4; ++nt)
                acc[mt][nt] = __builtin_amdgcn_wmma_f32_16x16x4_f32(
                    false, a, false, b[nt], (short)0, acc[mt][nt], false, false);
        }
    }

    // ---- epilogue 1: +b1, BatchNorm(eval), ReLU; stash 64x64 tile in LDS ----
    #pragma unroll
    for (int nt = 0; nt < 4; ++nt) {
        const int n  = nt * 16 + m;                  // feature index (per lane)
        const float sc = gamma[n] * rsqrtf(rvar[n] + BN_EPS);
        const float sh = beta[n] - rmean[n] * sc;
        const float bb = b1[n];
        #pragma unroll
        for (int mt = 0; mt < 4; ++mt)
            #pragma unroll
            for (int r = 0; r < 8; ++r) {
                const int mrow = mt * 16 + r + 8 * hi;
                float v = (acc[mt][nt][r] + bb) * sc + sh;
                lds[mrow * LDS_STRIDE + n] = fmaxf(v, 0.0f);
            }
    }
    __syncthreads();

    // ---- GEMM2: out = lds_supertile @ W2 ----
    #pragma unroll
    for (int mt = 0; mt < 4; ++mt)
        #pragma unroll
        for (int nt = 0; nt < 4; ++nt)
            acc[mt][nt] = (v8f){0,0,0,0,0,0,0,0};

    #pragma unroll
    for (int k = 0; k < DIM; k += 4) {
        const int ka = k + 2 * hi;
        v2f b[4];
        #pragma unroll
        for (int nt = 0; nt < 4; ++nt) {
            b[nt].x = W2[ka * DIM + nt * 16 + m];
            b[nt].y = W2[(ka + 1) * DIM + nt * 16 + m];
        }
        #pragma unroll
        for (int mt = 0; mt < 4; ++mt) {
            v2f a;
            a.x = lds[(mt * 16 + m) * LDS_STRIDE + ka];
            a.y = lds[(mt * 16 + m) * LDS_STRIDE + ka + 1];
            #pragma unroll
            for (int nt = 0; nt < 4; ++nt)
                acc[mt][nt] = __builtin_amdgcn_wmma_f32_16x16x4_f32(
                    false, a, false, b[nt], (short)0, acc[mt][nt], false, false);
        }
    }

    // ---- epilogue 2: +b2, ReLU, guarded store ----
    #pragma unroll
    for (int nt = 0; nt < 4; ++nt) {
        const int n  = nt * 16 + m;
        const float bb = b2[n];
        #pragma unroll
        for (int mt = 0; mt < 4; ++mt)
            #pragma unroll
            for (int r = 0; r < 8; ++r) {
                const int row = row0 + mt * 16 + r + 8 * hi;
                if (row < n_nodes)
                    Hout[(long long)row * DIM + n] = fmaxf(acc[mt][nt][r] + bb, 0.0f);
            }
    }
}

// ---------------------------------------------------------------------------
// Zero pooled buffer
// ---------------------------------------------------------------------------
__global__ void gin_zero(float* __restrict__ p, int n) {
    int t = blockIdx.x * blockDim.x + threadIdx.x;
    if (t < n) p[t] = 0.0f;
}

// ---------------------------------------------------------------------------
// global_add_pool: pooled[batch[i]] += h[i].  16 threads/node, float4 each.
// ---------------------------------------------------------------------------
__global__ void gin_pool(const float* __restrict__ H, const int* __restrict__ batch,
                         float* __restrict__ pooled, long long n_nodes) {
    long long t = (long long)blockIdx.x * blockDim.x + threadIdx.x;
    long long node = t >> 4;
    if (node >= n_nodes) return;
    int c = (int)(t & 15);
    int g = batch[node];
    const float4 v = *(const float4*)(H + node * DIM + c * 4);
    float* dp = pooled + (long long)g * DIM + c * 4;
    atomicAdd(dp + 0, v.x);
    atomicAdd(dp + 1, v.y);
    atomicAdd(dp + 2, v.z);
    atomicAdd(dp + 3, v.w);
}

// ---------------------------------------------------------------------------
// Head: out = relu(pooled@lin1 + b) @ lin2 + b2.  One 64-thread block / graph.
// ---------------------------------------------------------------------------
__global__ __launch_bounds__(64) void gin_head(
    const float* __restrict__ pooled,
    const float* __restrict__ l1w, const float* __restrict__ l1b,
    const float* __restrict__ l2w, const float* __restrict__ l2b,
    float* __restrict__ out) {
    __shared__ float pin[DIM];
    __shared__ float hidden[DIM];
    const int g = blockIdx.x;
    const int f = threadIdx.x;
    pin[f] = pooled[g * DIM + f];
    __syncthreads();
    float s = l1b[f];
    #pragma unroll 8
    for (int k = 0; k < DIM; ++k) s += pin[k] * l1w[k * DIM + f];
    hidden[f] = fmaxf(s, 0.0f);
    __syncthreads();
    if (f < OUTDIM) {
        float o = l2b[f];
        #pragma unroll 8
        for (int k = 0; k < DIM; ++k) o += hidden[k] * l2w[k * OUTDIM + f];
        out[g * OUTDIM + f] = o;
    }
}

// ---------------------------------------------------------------------------
extern "C" void kernel_launch(void* const* d_in, const int* in_sizes, int n_in,
                              void* d_out, int out_size, void* d_ws, size_t ws_size,
                              hipStream_t stream) {
    // setup_inputs order:
    // 0:x 1:edge_index 2:batch 3:W1 4:b1 5:gamma 6:beta 7:rmean 8:rvar 9:W2 10:b2
    // 11:lin1_w 12:lin1_b 13:lin2_w 14:lin2_b
    const float* x       = (const float*)d_in[0];
    const int*   eidx    = (const int*)d_in[1];
    const int*   batch   = (const int*)d_in[2];
    const float* W1      = (const float*)d_in[3];
    const float* b1      = (const float*)d_in[4];
    const float* gamma   = (const float*)d_in[5];
    const float* beta    = (const float*)d_in[6];
    const float* rmean   = (const float*)d_in[7];
    const float* rvar    = (const float*)d_in[8];
    const float* W2      = (const float*)d_in[9];
    const float* b2      = (const float*)d_in[10];
    const float* lin1_w  = (const float*)d_in[11];
    const float* lin1_b  = (const float*)d_in[12];
    const float* lin2_w  = (const float*)d_in[13];
    const float* lin2_b  = (const float*)d_in[14];
    float* out = (float*)d_out;

    const long long n_nodes = in_sizes[0] / DIM;
    const long long n_edges = in_sizes[1] / 2;
    const int*   esrc = eidx;              // edge_index[0, :]
    const int*   edst = eidx + n_edges;    // edge_index[1, :]

    // workspace layout: two ping-pong feature buffers + pooled
    float* buf0   = (float*)d_ws;
    float* buf1   = buf0 + n_nodes * DIM;
    float* pooled = buf1 + n_nodes * DIM;

    const long long n4 = n_nodes * DIM / 4;
    const int copyBlocks = 2048;
    const long long scatT = n_edges * 16;
    const int scatBlocks = (int)((scatT + 255) / 256);
    const int mlpBlocks = (int)((n_nodes + 63) / 64);   // one wave per 64-row supertile

    const float* hcur = x;
    for (int L = 0; L < 5; ++L) {
        float* agg = (L & 1) ? buf1 : buf0;
        gin_copy<<<copyBlocks, 256, 0, stream>>>((const float4*)hcur, (float4*)agg, n4);
        gin_scatter<<<scatBlocks, 256, 0, stream>>>(hcur, agg, esrc, edst, n_edges);
        gin_mlp<<<mlpBlocks, 32, 0, stream>>>(
            agg, agg,
            W1 + (long long)L * DIM * DIM, b1 + L * DIM,
            gamma + L * DIM, beta + L * DIM, rmean + L * DIM, rvar + L * DIM,
            W2 + (long long)L * DIM * DIM, b2 + L * DIM,
            (int)n_nodes);
        hcur = agg;
    }

    gin_zero<<<(NGRAPHS * DIM + 255) / 256, 256, 0, stream>>>(pooled, NGRAPHS * DIM);
    const long long poolT = n_nodes * 16;
    gin_pool<<<(int)((poolT + 255) / 256), 256, 0, stream>>>(hcur, batch, pooled, n_nodes);
    gin_head<<<NGRAPHS, 64, 0, stream>>>(pooled, lin1_w, lin1_b, lin2_w, lin2_b, out);
}